// ContextAttentionAdapterWrapper_58978490909116
// MI455X (gfx1250) — compile-verified
//
#include <hip/hip_runtime.h>
#include <hip/hip_bf16.h>
#include <math.h>

typedef float v2f __attribute__((ext_vector_type(2)));
typedef float v8f __attribute__((ext_vector_type(8)));

#define LN_THREADS 256

// ---------------------------------------------------------------------------
// block-wide sum reduction helper (256 threads)
// ---------------------------------------------------------------------------
__device__ __forceinline__ float block_reduce_sum(float v, float* red) {
  const int t = threadIdx.x;
  red[t] = v;
  __syncthreads();
  for (int o = 128; o > 0; o >>= 1) {
    if (t < o) red[t] += red[t + o];
    __syncthreads();
  }
  float r = red[0];
  __syncthreads();   // make red reusable
  return r;
}

// ---------------------------------------------------------------------------
// LayerNorm over rows; optionally also computes ep[row] = dot(raw_x, Wp)
// (residual path uses RAW embeddings, so the Wp-dot is on raw x).
// one block per row, 256 threads, D <= 3072.
// ---------------------------------------------------------------------------
__global__ __launch_bounds__(LN_THREADS)
void ln_rows_kernel(const float* __restrict__ X, const float* __restrict__ g,
                    const float* __restrict__ beta, float* __restrict__ Y, int D,
                    const float* __restrict__ Wp, float* __restrict__ ep) {
  __shared__ float xs[3072];
  __shared__ float red[LN_THREADS];
  const int row = blockIdx.x, t = threadIdx.x;
  const float* x = X + (size_t)row * D;

  float s = 0.f, wsum = 0.f;
  for (int d = t; d < D; d += LN_THREADS) {
    float v = x[d];
    xs[d] = v;
    s += v;
    if (Wp) wsum += v * Wp[d];
  }
  __syncthreads();
  const float mu = block_reduce_sum(s, red) / (float)D;

  float vs = 0.f;
  for (int d = t; d < D; d += LN_THREADS) {
    float dv = xs[d] - mu;
    vs += dv * dv;
  }
  const float var = block_reduce_sum(vs, red) / (float)D;
  const float rs = rsqrtf(var + 1e-5f);

  float* y = Y + (size_t)row * D;
  for (int d = t; d < D; d += LN_THREADS)
    y[d] = (xs[d] - mu) * rs * g[d] + beta[d];

  if (Wp) {
    float e = block_reduce_sum(wsum, red);
    if (t == 0) ep[row] = e;
  }
}

// ---------------------------------------------------------------------------
// WoWp[i] = dot(Wo[i,:], Wp)  (i < 512);  block 512: cb = dot(bo,Wp)+bp
// ---------------------------------------------------------------------------
__global__ __launch_bounds__(LN_THREADS)
void wowp_kernel(const float* __restrict__ Wo, const float* __restrict__ Wp,
                 const float* __restrict__ bo, const float* __restrict__ bp,
                 float* __restrict__ WoWp, float* __restrict__ cb) {
  __shared__ float red[LN_THREADS];
  const int i = blockIdx.x, t = threadIdx.x;
  const float* rowp = (i < 512) ? (Wo + (size_t)i * 3072) : bo;
  float s = 0.f;
  for (int d = t; d < 3072; d += LN_THREADS) s += rowp[d] * Wp[d];
  float r = block_reduce_sum(s, red);
  if (t == 0) {
    if (i < 512) WoWp[i] = r;
    else         cb[0]   = r + bp[0];
  }
}

// ---------------------------------------------------------------------------
// fp32 WMMA GEMM: C[M,N] = A[M,K] @ B[K,N], row-major, M%64==0, N%64==0, K%4==0
// block = (32,4): each wave owns a 16x64 output strip (A frag reused x4).
// V_WMMA_F32_16X16X4_F32 fragment layouts (ISA 7.12.2, 32-bit matrices):
//   A 16x4 : lane<16 -> M=lane, VGPR0/1 = K=k,k+1 ; lane>=16 -> K=k+2,k+3
//   B 4x16 : VGPR0 = row k+2*half, VGPR1 = row k+1+2*half, col = lane&15
//   D 16x16: VGPR r -> M = r + 8*half, N = lane&15
// ---------------------------------------------------------------------------
__global__ __launch_bounds__(128)
void gemm_f32_kernel(const float* __restrict__ A, const float* __restrict__ B,
                     float* __restrict__ C, int M, int N, int K) {
#if defined(__AMDGCN__) && __has_builtin(__builtin_amdgcn_wmma_f32_16x16x4_f32)
  const int lane = threadIdx.x;
  const int half = lane >> 4;       // 0 or 1
  const int l16  = lane & 15;
  const int m0 = (blockIdx.y * 4 + threadIdx.y) * 16;
  const int n0 = blockIdx.x * 64;

  v8f acc0 = {}, acc1 = {}, acc2 = {}, acc3 = {};
  const float* Ap  = A + (size_t)(m0 + l16) * K + half * 2;
  const float* Bp0 = B + n0 + l16;

  for (int k = 0; k < K; k += 4) {
    v2f a = *(const v2f*)(Ap + k);                       // A[m, k+2h], A[m, k+1+2h]
    const float* Bp = Bp0 + (size_t)(k + 2 * half) * N;
    v2f b0, b1, b2, b3;
    b0.x = Bp[0];   b0.y = Bp[(size_t)N + 0];
    b1.x = Bp[16];  b1.y = Bp[(size_t)N + 16];
    b2.x = Bp[32];  b2.y = Bp[(size_t)N + 32];
    b3.x = Bp[48];  b3.y = Bp[(size_t)N + 48];
    acc0 = __builtin_amdgcn_wmma_f32_16x16x4_f32(false, a, false, b0, (short)0, acc0, false, false);
    acc1 = __builtin_amdgcn_wmma_f32_16x16x4_f32(false, a, false, b1, (short)0, acc1, false, false);
    acc2 = __builtin_amdgcn_wmma_f32_16x16x4_f32(false, a, false, b2, (short)0, acc2, false, false);
    acc3 = __builtin_amdgcn_wmma_f32_16x16x4_f32(false, a, false, b3, (short)0, acc3, false, false);
  }

  float* Crow = C + (size_t)(m0 + half * 8) * N + n0 + l16;
  #pragma unroll
  for (int r = 0; r < 8; ++r) {
    Crow[(size_t)r * N + 0]  = acc0[r];
    Crow[(size_t)r * N + 16] = acc1[r];
    Crow[(size_t)r * N + 32] = acc2[r];
    Crow[(size_t)r * N + 48] = acc3[r];
  }
#else
#if defined(__AMDGCN__)
#warning "gfx1250 device pass: __builtin_amdgcn_wmma_f32_16x16x4_f32 unavailable -> scalar GEMM fallback"
#endif
  const int lane = threadIdx.x;
  const int m0 = (blockIdx.y * 4 + threadIdx.y) * 16;
  const int n0 = blockIdx.x * 64;
  for (int idx = lane; idx < 16 * 64; idx += 32) {
    const int r = idx >> 6, cc = idx & 63;
    float s = 0.f;
    const float* a = A + (size_t)(m0 + r) * K;
    const float* b = B + n0 + cc;
    for (int k = 0; k < K; ++k) s += a[k] * b[(size_t)k * N];
    C[(size_t)(m0 + r) * N + n0 + cc] = s;
  }
#endif
}

// ---------------------------------------------------------------------------
// vw1[c,h] = dot(v[c,h,:], WoWp[h*64:..])   (blocks 0..63)
// nvw[h]   = dot(null_v[h*64:..], WoWp[h*64:..])  (block 64)
// v[c,h,d] = KV[c*1024 + 512 + h*64 + d]
// ---------------------------------------------------------------------------
__global__ __launch_bounds__(256)
void prep_vw_kernel(const float* __restrict__ KV, const float* __restrict__ null_v,
                    const float* __restrict__ WoWp,
                    float* __restrict__ vw1, float* __restrict__ nvw) {
  const int c = blockIdx.x, t = threadIdx.x;
  const int h = t >> 5, lane = t & 31;
  const float* v = (c < 64) ? (KV + (size_t)c * 1024 + 512 + h * 64)
                            : (null_v + h * 64);
  const float* w = WoWp + h * 64;
  const int d = lane * 2;
  float s = v[d] * w[d] + v[d + 1] * w[d + 1];
  for (int o = 16; o > 0; o >>= 1) s += __shfl_down(s, o, 32);
  if (lane == 0) {
    if (c < 64) vw1[c * 8 + h] = s;
    else        nvw[h] = s;
  }
}

// ---------------------------------------------------------------------------
// Fused attention + folded-output kernel. One block per query position n.
//   logits: l0 = scale*q_h·null_k_h, l1 = scale*q_h·k[c,h] (masked)
//   pred[n,c] = softplus( ep[n] + cb + sum_h p0*nvw[h] + p1*vw1[c,h] )
// ---------------------------------------------------------------------------
__global__ __launch_bounds__(256)
void attn_out_kernel(const float* __restrict__ Q, const float* __restrict__ KV,
                     const float* __restrict__ null_k, const void* __restrict__ maskp,
                     const float* __restrict__ vw1, const float* __restrict__ nvw,
                     const float* __restrict__ ep, const float* __restrict__ cb,
                     float* __restrict__ out) {
  __shared__ float qs[512];
  __shared__ float sim0s[8];
  __shared__ float part[256];
  __shared__ int maskfmt;
  const int n = blockIdx.x, t = threadIdx.x;
  const float scale = 0.125f;  // 64^-0.5

  if (t == 0) {
    // bool mask may be shipped as int32 or uint8; an all-0/1 int32 view says
    // int format, bytes of a bool mask read as e.g. 0x01010101 say byte format.
    const int* mi = (const int*)maskp;
    int fmt = 1;
    for (int i = 0; i < 16; ++i) {
      int v = mi[i];
      if (v != 0 && v != 1) { fmt = 0; break; }
    }
    maskfmt = fmt;
  }
  qs[t]       = Q[(size_t)n * 512 + t];
  qs[t + 256] = Q[(size_t)n * 512 + 256 + t];
  __syncthreads();

  // null-key logits per head (wave h of 8)
  {
    const int h = t >> 5, lane = t & 31;
    const int base = h * 64, d = lane * 2;
    float s = qs[base + d] * null_k[base + d] + qs[base + d + 1] * null_k[base + d + 1];
    for (int o = 16; o > 0; o >>= 1) s += __shfl_down(s, o, 32);
    if (lane == 0) sim0s[h] = s * scale;
  }
  __syncthreads();

  // thread t: context c = t&63, head-pair hq = t>>6
  const int c = t & 63, hq = t >> 6;
  bool mval;
  if (maskfmt) mval = ((const int*)maskp)[c] != 0;
  else         mval = ((const unsigned char*)maskp)[c] != 0;

  float acc = 0.f;
  #pragma unroll
  for (int hh = 0; hh < 2; ++hh) {
    const int h = hq * 2 + hh, base = h * 64;
    const float* kp = KV + (size_t)c * 1024 + base;
    float s1 = 0.f;
    #pragma unroll 8
    for (int d = 0; d < 64; ++d) s1 += qs[base + d] * kp[d];
    const float l0 = sim0s[h];
    const float l1 = mval ? s1 * scale : -3.402823466e38f;
    const float m  = fmaxf(l0, l1);
    const float e0 = expf(l0 - m), e1 = expf(l1 - m);
    const float inv = 1.f / (e0 + e1);
    acc += (e0 * inv) * nvw[h] + (e1 * inv) * vw1[c * 8 + h];
  }
  part[t] = acc;
  __syncthreads();

  if (t < 64) {
    const float tot = part[t] + part[t + 64] + part[t + 128] + part[t + 192];
    const float z = ep[n] + tot + cb[0];
    const float sp = (z > 0.f) ? (z + log1pf(expf(-z))) : log1pf(expf(z));
    out[(size_t)n * 64 + t] = sp;  // layout (b=1, n, c)
  }
}

// ---------------------------------------------------------------------------
extern "C" void kernel_launch(void* const* d_in, const int* in_sizes, int n_in,
                              void* d_out, int out_size, void* d_ws, size_t ws_size,
                              hipStream_t stream) {
  const float* emb    = (const float*)d_in[0];   // (1, 896, 3072)
  const float* ctx    = (const float*)d_in[1];   // (64, 1, 2560)
  const void*  cmask  = d_in[2];                 // (64, 1) bool
  const float* qn_g   = (const float*)d_in[3];
  const float* qn_b   = (const float*)d_in[4];
  const float* kn_g   = (const float*)d_in[5];
  const float* kn_b   = (const float*)d_in[6];
  const float* Wq     = (const float*)d_in[7];   // (3072, 512)
  const float* Wkv    = (const float*)d_in[8];   // (2560, 1024)
  const float* null_k = (const float*)d_in[9];   // (512,)
  const float* null_v = (const float*)d_in[10];  // (512,)
  const float* Wo     = (const float*)d_in[11];  // (512, 3072)
  const float* bo     = (const float*)d_in[12];  // (3072,)
  const float* Wp     = (const float*)d_in[13];  // (3072, 1)
  const float* bp     = (const float*)d_in[14];  // (1,)
  float* out = (float*)d_out;

  // workspace layout (floats)
  float* ws   = (float*)d_ws;
  float* lnE  = ws;                    // 896*3072
  float* lnC  = lnE  + 896 * 3072;     // 64*2560
  float* Qm   = lnC  + 64 * 2560;      // 896*512
  float* KVm  = Qm   + 896 * 512;      // 64*1024
  float* WoWp = KVm  + 64 * 1024;      // 512
  float* epv  = WoWp + 512;            // 896
  float* cbv  = epv  + 896;            // 16 (1 used)
  float* vw1  = cbv  + 16;             // 64*8
  float* nvw  = vw1  + 512;            // 8

  // 1) LayerNorms (embeddings LN also folds ep[n] = raw_emb[n]·Wp)
  ln_rows_kernel<<<896, LN_THREADS, 0, stream>>>(emb, qn_g, qn_b, lnE, 3072, Wp, epv);
  ln_rows_kernel<<<64,  LN_THREADS, 0, stream>>>(ctx, kn_g, kn_b, lnC, 2560, nullptr, nullptr);

  // 2) WoWp = Wo@Wp (and cb = bo·Wp + bp)
  wowp_kernel<<<513, LN_THREADS, 0, stream>>>(Wo, Wp, bo, bp, WoWp, cbv);

  // 3) WMMA GEMMs: Q = lnE @ Wq (896x3072x512), KV = lnC @ Wkv (64x2560x1024)
  dim3 gb(32, 4);
  gemm_f32_kernel<<<dim3(512 / 64, 896 / 64), gb, 0, stream>>>(lnE, Wq,  Qm,  896, 512,  3072);
  gemm_f32_kernel<<<dim3(1024 / 64, 64 / 64), gb, 0, stream>>>(lnC, Wkv, KVm, 64,  1024, 2560);

  // 4) fold V through WoWp
  prep_vw_kernel<<<65, 256, 0, stream>>>(KVm, null_v, WoWp, vw1, nvw);

  // 5) fused attention + softplus output
  attn_out_kernel<<<896, 256, 0, stream>>>(Qm, KVm, null_k, cmask, vw1, nvw, epv, cbv, out);
}